// HopfLayer_35098472743222
// MI455X (gfx1250) — compile-verified
//
#include <hip/hip_runtime.h>
#include <hip/hip_bf16.h>

typedef __attribute__((ext_vector_type(2))) float v2f;
typedef __attribute__((ext_vector_type(8))) float v8f;

#define HOPF_DT    0.025f
#define HOPF_EPS   1e-05f

// Problem constants
#define NB   16
#define CI   3
#define CO   64
#define HH   224
#define WW   224
#define HW   (HH * WW)          // 50176
#define TILES (HW / 16)         // 3136 pixel-tiles of 16
#define OBLOCKS 4               // 64 / 16
#define CHUNKS 28               // pixel chunks per (b, oblock)
#define TILES_PER_CHUNK 112     // 3136 / 28
#define TILES_PER_WAVE 14       // 112 / 8 waves
#define K1_THREADS 256          // 8 wave32 waves

// d_ws float layout
#define WS_PSUM 0
#define WS_PSQ  (NB * OBLOCKS * CHUNKS * 16)            // 28672
#define WS_SCALE (2 * NB * OBLOCKS * CHUNKS * 16)       // 57344
#define WS_SHIFT (WS_SCALE + NB * CO)                   // 58368

// ---------------------------------------------------------------------------
// Kernel 1: u = W @ x via V_WMMA_F32_16X16X4_F32, then 8-step Hopf ODE,
// a = |z| -> d_out, plus deterministic per-chunk partial sums of a, a^2.
// ---------------------------------------------------------------------------
__global__ void __launch_bounds__(K1_THREADS)
hopf_k1(const float* __restrict__ x, const float* __restrict__ Wm,
        const float* __restrict__ omega, const float* __restrict__ gam,
        float* __restrict__ out, float* __restrict__ psum, float* __restrict__ psq)
{
    const int bid  = blockIdx.x;
    const int c    = bid % CHUNKS;
    const int ob   = (bid / CHUNKS) % OBLOCKS;
    const int b    = bid / (CHUNKS * OBLOCKS);

    const int lane = threadIdx.x & 31;
    const int wave = threadIdx.x >> 5;
    const int hi   = lane >> 4;        // 0: lanes 0-15 (K=0,1 / M=r), 1: lanes 16-31 (K=2,pad / M=r+8)
    const int ln   = lane & 15;

    // --- A operand: 16x4 tile of W (K=3 padded to 4). ISA 32-bit A layout:
    // lanes 0-15 hold (K0,K1) for row M=lane; lanes 16-31 hold (K2,K3) for row M=lane-16.
    const int o_a = ob * 16 + ln;
    v2f Aop;
    if (hi) { Aop.x = Wm[o_a * 3 + 2]; Aop.y = 0.0f; }
    else    { Aop.x = Wm[o_a * 3 + 0]; Aop.y = Wm[o_a * 3 + 1]; }

    // Per-lane channel params: VGPR r of the C/D tile is o = ob*16 + r + 8*hi
    float om[8], ga[8];
#pragma unroll
    for (int r = 0; r < 8; ++r) {
        const int o_r = ob * 16 + hi * 8 + r;
        om[r] = omega[o_r];
        ga[r] = gam[o_r];
    }

    const float* xb = x + (size_t)b * CI * HW;
    float sum[8], sq[8];
#pragma unroll
    for (int r = 0; r < 8; ++r) { sum[r] = 0.0f; sq[r] = 0.0f; }

    for (int i = 0; i < TILES_PER_WAVE; ++i) {
        const int tile = c * TILES_PER_CHUNK + wave * TILES_PER_WAVE + i;
        const int pix  = tile * 16 + ln;

        // --- B operand: 4x16 tile of x (K rows x pixel cols), same striping as A.
        v2f Bop;
        if (hi) { Bop.x = xb[2 * HW + pix]; Bop.y = 0.0f; }
        else    { Bop.x = xb[pix];          Bop.y = xb[HW + pix]; }

        v8f acc = {};
        acc = __builtin_amdgcn_wmma_f32_16x16x4_f32(
            /*neg_a=*/false, Aop, /*neg_b=*/false, Bop,
            /*c_mod=*/(short)0, acc, /*reuse_a=*/false, /*reuse_b=*/false);

#pragma unroll
        for (int r = 0; r < 8; ++r) {
            const float u = acc[r];
            // step 1 from z=0 collapses to zr = DT*u, zi = 0
            float zr = HOPF_DT * u;
            float zi = 0.0f;
#pragma unroll
            for (int t = 0; t < 7; ++t) {
                const float r2  = zr * zr + zi * zi;
                const float cc  = ga[r] - r2;
                const float dzr = cc * zr - om[r] * zi;
                const float dzi = cc * zi + om[r] * zr;
                zr = zr + HOPF_DT * (dzr + u);
                zi = zi + HOPF_DT * dzi;
            }
            const float a = sqrtf(zr * zr + zi * zi);
            const int o_r = ob * 16 + hi * 8 + r;
            out[((size_t)(b * CO + o_r)) * HW + pix] = a;
            sum[r] += a;
            sq[r]  += a * a;
        }
    }

    // --- reduce each half-wave (16 lanes share one o per r) via shfl-xor
    __shared__ float s_sum[8 * 16];
    __shared__ float s_sq[8 * 16];
#pragma unroll
    for (int r = 0; r < 8; ++r) {
        float vs = sum[r], vq = sq[r];
        for (int m = 8; m >= 1; m >>= 1) {
            vs += __shfl_xor(vs, m, 32);
            vq += __shfl_xor(vq, m, 32);
        }
        if (ln == 0) {
            s_sum[wave * 16 + hi * 8 + r] = vs;
            s_sq [wave * 16 + hi * 8 + r] = vq;
        }
    }
    __syncthreads();

    // --- cross-wave reduction, one partial per (b, ob, chunk, ro)
    if (threadIdx.x < 16) {
        float ts = 0.0f, tq = 0.0f;
#pragma unroll
        for (int w2 = 0; w2 < 8; ++w2) {
            ts += s_sum[w2 * 16 + threadIdx.x];
            tq += s_sq [w2 * 16 + threadIdx.x];
        }
        const int pidx = ((b * OBLOCKS + ob) * CHUNKS + c) * 16 + threadIdx.x;
        psum[pidx] = ts;
        psq [pidx] = tq;
    }
}

// ---------------------------------------------------------------------------
// Kernel 2: finalize per-(b,o) stats -> scale/shift
// ---------------------------------------------------------------------------
__global__ void hopf_k2(const float* __restrict__ psum, const float* __restrict__ psq,
                        const float* __restrict__ ng, const float* __restrict__ nb,
                        float* __restrict__ scale, float* __restrict__ shift)
{
    const int bo = blockIdx.x * blockDim.x + threadIdx.x;
    if (bo >= NB * CO) return;
    const int b  = bo / CO;
    const int o  = bo % CO;
    const int ob = o / 16;
    const int ro = o % 16;

    float ts = 0.0f, tq = 0.0f;
    const int base = (b * OBLOCKS + ob) * CHUNKS * 16 + ro;
    for (int c = 0; c < CHUNKS; ++c) {
        ts += psum[base + c * 16];
        tq += psq [base + c * 16];
    }
    const float inv_n = 1.0f / (float)HW;
    const float mean  = ts * inv_n;
    float var = tq * inv_n - mean * mean;
    var = var > 0.0f ? var : 0.0f;
    const float sc = ng[o] * rsqrtf(var + HOPF_EPS);
    scale[bo] = sc;
    shift[bo] = nb[o] - mean * sc;
}

// ---------------------------------------------------------------------------
// Kernel 3: in-place normalize, float4-vectorized. Plane = 50176 = 49*1024,
// so each 256-thread block lies entirely in one (b,o) plane (uniform bo).
// ---------------------------------------------------------------------------
__global__ void __launch_bounds__(256)
hopf_k3(float* __restrict__ out, const float* __restrict__ scale,
        const float* __restrict__ shift)
{
    const size_t tid = (size_t)blockIdx.x * blockDim.x + threadIdx.x;
    const size_t e   = tid * 4;
    const int bo = (int)(e / HW);
    const float sc = scale[bo];
    const float sh = shift[bo];
    float4* p = reinterpret_cast<float4*>(out + e);
    float4 v = *p;
    v.x = v.x * sc + sh;
    v.y = v.y * sc + sh;
    v.z = v.z * sc + sh;
    v.w = v.w * sc + sh;
    *p = v;
}

// ---------------------------------------------------------------------------
extern "C" void kernel_launch(void* const* d_in, const int* in_sizes, int n_in,
                              void* d_out, int out_size, void* d_ws, size_t ws_size,
                              hipStream_t stream) {
    const float* x     = (const float*)d_in[0];
    const float* Wm    = (const float*)d_in[1];
    const float* omeg  = (const float*)d_in[2];
    const float* gam   = (const float*)d_in[3];
    const float* ng    = (const float*)d_in[4];
    const float* nbb   = (const float*)d_in[5];
    float* out = (float*)d_out;

    float* wsf   = (float*)d_ws;
    float* psum  = wsf + WS_PSUM;
    float* psq   = wsf + WS_PSQ;
    float* scale = wsf + WS_SCALE;
    float* shift = wsf + WS_SHIFT;

    const int k1_blocks = NB * OBLOCKS * CHUNKS;                 // 1792
    hopf_k1<<<k1_blocks, K1_THREADS, 0, stream>>>(x, Wm, omeg, gam, out, psum, psq);

    hopf_k2<<<(NB * CO + 255) / 256, 256, 0, stream>>>(psum, psq, ng, nbb, scale, shift);

    const int k3_blocks = (NB * CO * HW / 4) / 256;              // 50176
    hopf_k3<<<k3_blocks, 256, 0, stream>>>(out, scale, shift);
}